// InterGraphOperation_11871289606433
// MI455X (gfx1250) — compile-verified
//
#include <hip/hip_runtime.h>
#include <hip/hip_bf16.h>

// Problem constants (from reference): B=4, R=64, F=1024, D=128
#define BB 4
#define RR 64
#define FF 1024
#define DD 128

typedef __attribute__((ext_vector_type(2))) float v2f;
typedef __attribute__((ext_vector_type(8))) float v8f;

// ---------------------------------------------------------------------------
// Kernel 0: w_d[d] = sum_k W1d[k][d]  (column sums of the third D x D block)
// ---------------------------------------------------------------------------
__global__ __launch_bounds__(DD) void colsum_kernel(const float* __restrict__ W1d,
                                                    float* __restrict__ wd) {
    int d = threadIdx.x;
    float s = 0.f;
    #pragma unroll 8
    for (int k = 0; k < DD; ++k) s += W1d[k * DD + d];
    wd[d] = s;
}

// ---------------------------------------------------------------------------
// Generic WMMA f32 GEMM:  C[b] = op( A[b] (MxK) @ Bm[b] (KxN) + bias [+ C] )
// One wave computes a 16x16 tile via V_WMMA_F32_16X16X4_F32, K-loop step 4.
// A-frag:  lane l -> row m0+(l&15), float2 at col k+2*(l>>4)      (1 b64 load)
// B-frag:  lane l -> col n0+(l&15), rows k+2*(l>>4), k+2*(l>>4)+1 (2 b32 loads)
// C/D:     vgpr i -> row m0+i+8*(l>>4), col n0+(l&15)
// Batch via blockIdx.y with element strides. M,N assumed multiples of 16.
// ---------------------------------------------------------------------------
__global__ __launch_bounds__(128) void wmma_gemm_f32_kernel(
    const float* __restrict__ A,  int lda, int strideA,
    const float* __restrict__ Bm, int ldb, int strideB,
    float* __restrict__ C,        int ldc, int strideC,
    int M, int N, int K,
    const float* __restrict__ bias, int accumulate, int relu) {

    const int tN   = N >> 4;
    const int tM   = M >> 4;
    const int tile = blockIdx.x * blockDim.y + threadIdx.y;
    if (tile >= tM * tN) return;                 // whole wave exits together

    const int m0   = (tile / tN) << 4;
    const int n0   = (tile % tN) << 4;
    const int lane = threadIdx.x;
    const int r    = lane & 15;
    const int hi   = lane >> 4;

    const int batch = blockIdx.y;
    const float* Ab = A  + (size_t)batch * (size_t)strideA;
    const float* Bb = Bm + (size_t)batch * (size_t)strideB;
    float*       Cb = C  + (size_t)batch * (size_t)strideC;

    const float* aptr = Ab + (size_t)(m0 + r) * lda + 2 * hi;
    const float* bptr = Bb + (size_t)(2 * hi) * ldb + n0 + r;

    v8f c = {};
    for (int k = 0; k < K; k += 4) {
        v2f a = *reinterpret_cast<const v2f*>(aptr + k);
        v2f b;
        b.x = bptr[(size_t)k * ldb];
        b.y = bptr[(size_t)(k + 1) * ldb];
        // D = A*B + C  (8 args: neg_a, A, neg_b, B, c_mod, C, reuse_a, reuse_b)
        c = __builtin_amdgcn_wmma_f32_16x16x4_f32(false, a, false, b,
                                                  (short)0, c, false, false);
    }

    const int n  = n0 + r;
    const float bv = bias ? bias[n] : 0.0f;
    #pragma unroll
    for (int i = 0; i < 8; ++i) {
        const int m = m0 + i + 8 * hi;
        float v = c[i] + bv;
        if (accumulate) v += Cb[(size_t)m * ldc + n];
        if (relu)       v  = fmaxf(v, 0.0f);
        Cb[(size_t)m * ldc + n] = v;
    }
}

// ---------------------------------------------------------------------------
// Fused score + softmax kernel. One 256-thread block per (b, r).
//   score[f] = relu(b2 + sum_d relu(rp[d] + fp[f,d] + g*wd[d]) * W2[d])
//   edge[f]  = softmax_f(score)
// rp already has b1 folded in (bias of the r_proj GEMM).
// ---------------------------------------------------------------------------
__global__ __launch_bounds__(256) void score_softmax_kernel(
    const float* __restrict__ rproj,   // (B*R, D)  with b1 folded
    const float* __restrict__ fproj,   // (B*F, D)
    const float* __restrict__ geo,     // (B*R, F)
    const float* __restrict__ wd,      // (D)
    const float* __restrict__ W2,      // (D)
    const float* __restrict__ b2p,     // (1)
    float* __restrict__ edge) {        // (B*R, F)

    const int br  = blockIdx.x;        // 0 .. B*R-1
    const int b   = br >> 6;           // R = 64
    const int tid = threadIdx.x;

    __shared__ float s_rp[DD];
    __shared__ float s_wd[DD];
    __shared__ float s_w2[DD];
    __shared__ float s_red[256];

    if (tid < DD) {
        s_rp[tid] = rproj[br * DD + tid];
        s_wd[tid] = wd[tid];
        s_w2[tid] = W2[tid];
    }
    __syncthreads();

    const float b2 = b2p[0];
    float sc[4];
    #pragma unroll
    for (int j = 0; j < 4; ++j) {
        const int f = tid + j * 256;
        const float g = geo[br * FF + f];
        const float* fp = fproj + (size_t)(b * FF + f) * DD;
        float acc = 0.f;
        #pragma unroll 4
        for (int d = 0; d < DD; d += 4) {
            const float4 v = *reinterpret_cast<const float4*>(fp + d);
            acc += fmaxf(fmaf(g, s_wd[d + 0], s_rp[d + 0] + v.x), 0.f) * s_w2[d + 0];
            acc += fmaxf(fmaf(g, s_wd[d + 1], s_rp[d + 1] + v.y), 0.f) * s_w2[d + 1];
            acc += fmaxf(fmaf(g, s_wd[d + 2], s_rp[d + 2] + v.z), 0.f) * s_w2[d + 2];
            acc += fmaxf(fmaf(g, s_wd[d + 3], s_rp[d + 3] + v.w), 0.f) * s_w2[d + 3];
        }
        sc[j] = fmaxf(acc + b2, 0.f);
    }

    // block-wide max
    float lmax = fmaxf(fmaxf(sc[0], sc[1]), fmaxf(sc[2], sc[3]));
    s_red[tid] = lmax;
    __syncthreads();
    for (int s = 128; s > 0; s >>= 1) {
        if (tid < s) s_red[tid] = fmaxf(s_red[tid], s_red[tid + s]);
        __syncthreads();
    }
    const float m = s_red[0];
    __syncthreads();

    // exp + block-wide sum
    float e[4], lsum = 0.f;
    #pragma unroll
    for (int j = 0; j < 4; ++j) { e[j] = __expf(sc[j] - m); lsum += e[j]; }
    s_red[tid] = lsum;
    __syncthreads();
    for (int s = 128; s > 0; s >>= 1) {
        if (tid < s) s_red[tid] += s_red[tid + s];
        __syncthreads();
    }
    const float inv = 1.0f / s_red[0];

    #pragma unroll
    for (int j = 0; j < 4; ++j)
        edge[(size_t)br * FF + tid + j * 256] = e[j] * inv;
}

// ---------------------------------------------------------------------------
// Host-side launch helper for the WMMA GEMM
// ---------------------------------------------------------------------------
static inline void launch_gemm(const float* A, int lda, int sA,
                               const float* Bm, int ldb, int sB,
                               float* C, int ldc, int sC,
                               int M, int N, int K,
                               const float* bias, int accumulate, int relu,
                               int batches, hipStream_t stream) {
    const int tiles  = (M >> 4) * (N >> 4);
    const int wavesPerBlock = 4;
    dim3 block(32, wavesPerBlock);
    dim3 grid((tiles + wavesPerBlock - 1) / wavesPerBlock, batches);
    wmma_gemm_f32_kernel<<<grid, block, 0, stream>>>(
        A, lda, sA, Bm, ldb, sB, C, ldc, sC, M, N, K, bias, accumulate, relu);
}

extern "C" void kernel_launch(void* const* d_in, const int* in_sizes, int n_in,
                              void* d_out, int out_size, void* d_ws, size_t ws_size,
                              hipStream_t stream) {
    (void)in_sizes; (void)n_in; (void)out_size; (void)ws_size;

    const float* robot    = (const float*)d_in[0]; // (B,R,D)
    const float* frontier = (const float*)d_in[1]; // (B,F,D)
    const float* geo      = (const float*)d_in[2]; // (B,R,F)
    const float* W1       = (const float*)d_in[3]; // (3D,D)
    const float* b1       = (const float*)d_in[4]; // (D)
    const float* W2       = (const float*)d_in[5]; // (D,1)
    const float* b2       = (const float*)d_in[6]; // (1)
    const float* Wn       = (const float*)d_in[7]; // (2D,D)
    const float* bn       = (const float*)d_in[8]; // (D)

    float* out  = (float*)d_out;
    float* newr = out;                       // (B,R,D)   = 32768 floats
    float* edge = out + BB * RR * DD;        // (B,R,F)   = 262144 floats

    // Workspace layout (f32): wd | r_proj | f_proj | messages  (~2.4 MB)
    float* ws    = (float*)d_ws;
    float* wd    = ws;                                   // 128
    float* rproj = wd + DD;                              // B*R*D
    float* fproj = rproj + BB * RR * DD;                 // B*F*D
    float* msgs  = fproj + BB * FF * DD;                 // B*R*D

    const float* W1r = W1;                 // rows [0, D)
    const float* W1f = W1 + DD * DD;       // rows [D, 2D)
    const float* W1d = W1 + 2 * DD * DD;   // rows [2D, 3D)
    const float* WnT = Wn;                 // rows [0, D)   (robot part)
    const float* WnB = Wn + DD * DD;       // rows [D, 2D)  (messages part)

    // 1) w_d = colsum(W1d)
    colsum_kernel<<<1, DD, 0, stream>>>(W1d, wd);

    // 2) r_proj = robot @ W1r + b1        (M=256, N=128, K=128)
    launch_gemm(robot, DD, 0, W1r, DD, 0, rproj, DD, 0,
                BB * RR, DD, DD, b1, 0, 0, 1, stream);

    // 3) f_proj = frontier @ W1f          (M=4096, N=128, K=128)
    launch_gemm(frontier, DD, 0, W1f, DD, 0, fproj, DD, 0,
                BB * FF, DD, DD, nullptr, 0, 0, 1, stream);

    // 4) fused scores + softmax -> edge_weights (directly into d_out)
    score_softmax_kernel<<<BB * RR, 256, 0, stream>>>(
        rproj, fproj, geo, wd, W2, b2, edge);

    // 5) messages = edge @ frontier       (batched: M=64, N=128, K=1024)
    launch_gemm(edge, FF, RR * FF, frontier, DD, FF * DD, msgs, DD, RR * DD,
                RR, DD, FF, nullptr, 0, 0, BB, stream);

    // 6) new = relu(robot @ Wn_top + bn + messages @ Wn_bot)
    launch_gemm(robot, DD, 0, WnT, DD, 0, newr, DD, 0,
                BB * RR, DD, DD, bn, 0, 0, 1, stream);
    launch_gemm(msgs, DD, 0, WnB, DD, 0, newr, DD, 0,
                BB * RR, DD, DD, nullptr, 1, 1, 1, stream);
}